// functional_square_loss_23390391894487
// MI455X (gfx1250) — compile-verified
//
#include <hip/hip_runtime.h>

typedef float v4f __attribute__((ext_vector_type(4)));
typedef int   v4i __attribute__((ext_vector_type(4)));
typedef float v2f __attribute__((ext_vector_type(2)));
typedef float v8f __attribute__((ext_vector_type(8)));

#define MARGIN_F 1.0f
#define TPB      256
#define NBLK_MAX 8192
#define NCOMP    6
#define ROWW     16   // padded partial-row width (floats) for WMMA B tiles

// ---------------------------------------------------------------------------
// Kernel 1: streaming masked reductions. HBM-bound: 268 MB @ 23.3 TB/s.
// b128 non-temporal loads, 6 accumulators/thread, wave32 shfl_xor tree,
// LDS cross-wave combine, one padded 16-float row of partials per block.
// ---------------------------------------------------------------------------
__global__ __launch_bounds__(TPB) void hinge2_partials(
    const float* __restrict__ preds, const int* __restrict__ labels,
    float* __restrict__ ws, int nv4)
{
    const v4f* pv = (const v4f*)preds;
    const v4i* lv = (const v4i*)labels;
    const int tid    = threadIdx.x;
    const int gid    = blockIdx.x * TPB + tid;
    const int stride = gridDim.x * TPB;

    float cp = 0.f, sz = 0.f, szz = 0.f;   // positives: count, sum z, sum z^2
    float cn = 0.f, sp = 0.f, spp = 0.f;   // negatives: count, sum p, sum p^2

    for (int i = gid; i < nv4; i += stride) {
        v4f p = __builtin_nontemporal_load(&pv[i]);
        v4i l = __builtin_nontemporal_load(&lv[i]);
#pragma unroll
        for (int k = 0; k < 4; ++k) {
            float pk  = p[k];
            bool  pos = (l[k] == 1);
            float z   = MARGIN_F - pk;
            if (pos) { cp += 1.0f; sz += z;  szz = fmaf(z, z, szz);  }
            else     { cn += 1.0f; sp += pk; spp = fmaf(pk, pk, spp); }
        }
    }

    // wave32 butterfly reduction
#pragma unroll
    for (int m = 16; m >= 1; m >>= 1) {
        cp  += __shfl_xor(cp,  m, 32);
        sz  += __shfl_xor(sz,  m, 32);
        szz += __shfl_xor(szz, m, 32);
        cn  += __shfl_xor(cn,  m, 32);
        sp  += __shfl_xor(sp,  m, 32);
        spp += __shfl_xor(spp, m, 32);
    }

    __shared__ float lds[TPB / 32][NCOMP];
    const int wave = tid >> 5, lane = tid & 31;
    if (lane == 0) {
        lds[wave][0] = cp;  lds[wave][1] = sz;  lds[wave][2] = szz;
        lds[wave][3] = cn;  lds[wave][4] = sp;  lds[wave][5] = spp;
    }
    __syncthreads();

    // threads 0..15 emit one padded row: [a, sz, szz, n, sp, spp, 0 ... 0]
    if (tid < ROWW) {
        float v = 0.0f;
        if (tid < NCOMP) {
#pragma unroll
            for (int w = 0; w < TPB / 32; ++w) v += lds[w][tid];
        }
        ws[blockIdx.x * ROWW + tid] = v;
    }
}

// ---------------------------------------------------------------------------
// Kernel 2: single wave32. Column-sum of the [nrows x 16] partial matrix via
// V_WMMA_F32_16X16X4_F32 with A = ones(16x4):  D = 1 * B + C accumulates the
// per-column sums (replicated across M). A = all-ones makes the result
// invariant to K-permutation in the operand layouts; columns are lane%16.
// Then a 5-flop combine on lane 0.
// ---------------------------------------------------------------------------
__global__ __launch_bounds__(32) void hinge2_final(
    const float* __restrict__ ws, float* __restrict__ out, int nrows)
{
    const int lane = threadIdx.x;        // 0..31, EXEC all ones (WMMA req.)
    const int col  = lane & 15;
    const int half = lane >> 4;          // which pair of K-rows this lane holds

    v8f acc = {};                        // C/D accumulator: 16x16 f32
    v2f ones; ones.x = 1.0f; ones.y = 1.0f;   // A = ones(16x4)

    for (int r = 0; r < nrows; r += 4) {
        v2f b;                           // B tile: rows r..r+3, 16 columns
        b.x = ws[(r + 2 * half + 0) * ROWW + col];
        b.y = ws[(r + 2 * half + 1) * ROWW + col];
        acc = __builtin_amdgcn_wmma_f32_16x16x4_f32(
                  /*neg_a=*/false, ones, /*neg_b=*/false, b,
                  /*c_mod=*/(short)0, acc, /*reuse_a=*/false, /*reuse_b=*/false);
    }

    // D vgpr0, lane j (0..15) = column-sum j (M=0 row; all rows identical)
    float v    = acc[0];
    float a    = __shfl(v, 0, 32);
    float szS  = __shfl(v, 1, 32);
    float szzS = __shfl(v, 2, 32);
    float cnS  = __shfl(v, 3, 32);
    float spS  = __shfl(v, 4, 32);
    float sppS = __shfl(v, 5, 32);

    if (lane == 0)
        out[0] = a * sppS + 2.0f * szS * spS + szzS * cnS;
}

// ---------------------------------------------------------------------------
extern "C" void kernel_launch(void* const* d_in, const int* in_sizes, int n_in,
                              void* d_out, int out_size, void* d_ws, size_t ws_size,
                              hipStream_t stream)
{
    const float* preds  = (const float*)d_in[0];
    const int*   labels = (const int*)d_in[1];
    float*       out    = (float*)d_out;
    float*       ws     = (float*)d_ws;

    const int n   = in_sizes[0];
    const int nv4 = n >> 2;             // N is 2^25, divisible by 4

    // Deterministic: ws_size is fixed by the harness. Cap partial rows by
    // available workspace; keep a multiple of 4 rows for the WMMA tiles.
    size_t rows_fit = ws_size / (ROWW * sizeof(float));
    int nblk = NBLK_MAX;
    if ((size_t)nblk > rows_fit) nblk = (int)rows_fit;
    nblk &= ~3;
    if (nblk < 4) nblk = 4;

    hinge2_partials<<<nblk, TPB, 0, stream>>>(preds, labels, ws, nv4);
    hinge2_final<<<1, 32, 0, stream>>>(ws, out, nblk);
}